// PSRoIAlign_43267500540390
// MI455X (gfx1250) — compile-verified
//
#include <hip/hip_runtime.h>
#include <stdint.h>

// PS-RoI-Align, MI455X (gfx1250, wave32).
// N=4, C=784=16*7*7, H=W=112, K ROIs, P=7, SR=2, SCALE=1/16.
// One thread = one (roi, bin) position x 4 of the 16 output channels:
//   - sample/bilinear math computed once, reused for 4 channels
//   - x-adjacent corner pairs fetched as single 8B gathers: 32 b64 loads/thread
// ROI rows needed by the block are staged into LDS with CDNA5 async-to-LDS DMA.

#define HGT     112
#define WID     112
#define COUT    16
#define CIN     784           // COUT * 49
#define PLANE   (HGT * WID)   // 12544
#define RSCALE  0.0625f

// 8-byte pair with only 4-byte alignment guarantee (xb may be odd)
struct __attribute__((aligned(4))) fpair { float a, b; };

__global__ __launch_bounds__(256) void psroi_align_kernel(
    const float* __restrict__ feat,   // [N, 784, 112, 112]
    const float* __restrict__ rois,   // [K, 5]
    float* __restrict__ out,          // [K, 16, 7, 7]
    int K)
{
    __shared__ float s_roi[16];       // up to 3 ROI rows (15 floats) + pad

    const int lane = threadIdx.x;
    const int tid  = blockIdx.x * 256 + lane;

    // ---- Stage the <=3 ROI rows this block touches into LDS (async DMA) ----
    // Block covers 64 positions: pos0 .. pos0+63  ->  k in [k_first, k_first+2]
    const int pos0    = (blockIdx.x * 256) >> 2;
    const int k_first = pos0 / 49;
    if (lane < 15) {
        // destination LDS byte offset: low 32 bits of generic pointer == LDS offset
        unsigned lds_off = (unsigned)(size_t)(&s_roi[lane]);
        int gidx = k_first * 5 + lane;
        int gmax = K * 5 - 1;
        if (gidx > gmax) gidx = gmax;                 // clamp OOB reads
        unsigned goff = (unsigned)gidx * 4u;          // byte offset (GVS mode)
        asm volatile("global_load_async_to_lds_b32 %0, %1, %2"
                     :: "v"(lds_off), "v"(goff), "s"(rois)
                     : "memory");
    }
    asm volatile("s_wait_asynccnt 0" ::: "memory");
    __syncthreads();

    const int pos  = tid >> 2;        // roi*49 + ph*7 + pw
    const int quad = tid & 3;         // which group of 4 channels
    if (pos >= K * 49) return;

    const int k  = pos / 49;
    const int pp = pos - k * 49;      // ph*7 + pw
    const int ph = pp / 7;
    const int pw = pp - ph * 7;

    // ---- ROI geometry (from LDS) ----
    const float* r = &s_roi[(k - k_first) * 5];
    const int   b  = (int)r[0];
    const float x1 = r[1] * RSCALE - 0.5f;
    const float y1 = r[2] * RSCALE - 0.5f;
    const float roi_w = fmaxf(r[3] * RSCALE - 0.5f - x1, 0.1f);
    const float roi_h = fmaxf(r[4] * RSCALE - 0.5f - y1, 0.1f);
    const float bsw = roi_w * (1.0f / 7.0f);
    const float bsh = roi_h * (1.0f / 7.0f);

    // ---- 2x2 samples -> 8 pair-offsets + 16 weights, branchless validity ----
    // Per sample: pair at row yl (cols xb,xb+1) and pair at row yh.
    int   offL[4], offH[4];           // element offsets of the row-pairs
    float wts[16];                    // [j*4+0..3] = wLa, wLb, wHa, wHb
    #pragma unroll
    for (int iy = 0; iy < 2; ++iy) {
        const float yq = y1 + ((float)ph + 0.25f + 0.5f * (float)iy) * bsh;
        const bool  vy = (yq >= -1.0f) && (yq <= (float)HGT);
        const float yc = fminf(fmaxf(yq, 0.0f), (float)(HGT - 1));
        const int   yl = (int)yc;                       // floor, yc >= 0
        const int   yh = min(yl + 1, HGT - 1);
        const float ly = yc - (float)yl;
        const float hy = 1.0f - ly;
        #pragma unroll
        for (int ix = 0; ix < 2; ++ix) {
            const float xq = x1 + ((float)pw + 0.25f + 0.5f * (float)ix) * bsw;
            const bool  vx = (xq >= -1.0f) && (xq <= (float)WID);
            const float xc = fminf(fmaxf(xq, 0.0f), (float)(WID - 1));
            const int   xl = (int)xc;
            const float lx = xc - (float)xl;
            const float hx = 1.0f - lx;
            const float vm = (vy && vx) ? 1.0f : 0.0f;  // zero weights if invalid
            // pair base column; right-edge clamp folds weight onto 2nd element
            const bool  in  = (xl < WID - 1);
            const int   xb  = in ? xl : (WID - 2);
            const float wa  = in ? (vm * hx) : 0.0f;    // col xb
            const float wb  = in ? (vm * lx) : vm;      // col xb+1 (edge: hx==1,lx==0)
            const int j = iy * 2 + ix;
            offL[j] = yl * WID + xb;
            offH[j] = yh * WID + xb;
            wts[j * 4 + 0] = hy * wa;
            wts[j * 4 + 1] = hy * wb;
            wts[j * 4 + 2] = ly * wa;
            wts[j * 4 + 3] = ly * wb;
        }
    }

    // ---- Gather + weighted sum for 4 channels (32 independent b64 loads) ----
    const int cout_base = quad * 4;
    const float* plane = feat + ((size_t)b * CIN + (size_t)cout_base * 49 + (size_t)pp) * PLANE;
    float*       op    = out  + ((size_t)k * COUT + (size_t)cout_base) * 49 + (size_t)pp;

    #pragma unroll
    for (int c = 0; c < 4; ++c) {
        const float* pl = plane + (size_t)c * (49 * PLANE);
        float acc = 0.0f;
        #pragma unroll
        for (int j = 0; j < 4; ++j) {
            const fpair pL = *(const fpair*)(pl + offL[j]);   // row yl: cols xb, xb+1
            const fpair pH = *(const fpair*)(pl + offH[j]);   // row yh: cols xb, xb+1
            acc = fmaf(wts[j * 4 + 0], pL.a, acc);
            acc = fmaf(wts[j * 4 + 1], pL.b, acc);
            acc = fmaf(wts[j * 4 + 2], pH.a, acc);
            acc = fmaf(wts[j * 4 + 3], pH.b, acc);
        }
        op[(size_t)c * 49] = acc * 0.25f;                     // mean over 2x2 samples
    }
}

extern "C" void kernel_launch(void* const* d_in, const int* in_sizes, int n_in,
                              void* d_out, int out_size, void* d_ws, size_t ws_size,
                              hipStream_t stream)
{
    const float* feat = (const float*)d_in[0];
    const float* rois = (const float*)d_in[1];
    float*       out  = (float*)d_out;

    const int K       = in_sizes[1] / 5;      // 1024
    const int total   = K * 49 * 4;           // one thread per (pos, 4-chan group)
    const int nblocks = (total + 255) / 256;  // 784 blocks of 8 wave32s

    psroi_align_kernel<<<nblocks, 256, 0, stream>>>(feat, rois, out, K);
}